// DAMD_56435870269721
// MI455X (gfx1250) — compile-verified
//
#include <hip/hip_runtime.h>
#include <stdint.h>

// ---------------- problem constants (match reference) ----------------
namespace {
constexpr int kB    = 8;      // batch
constexpr int kT    = 256;    // TDEC
constexpr int kV    = 16000;  // vocab
constexpr int kJ    = 256;    // T1 == T2 (copy positions)
constexpr int kVOOV = 16256;  // V + 256
constexpr int kNOOV = 256;
}

typedef __attribute__((ext_vector_type(2))) float v2f;
typedef __attribute__((ext_vector_type(8))) float v8f;
typedef __attribute__((ext_vector_type(4))) int   v4i;

// ---------------- CDNA5 feature probes (device pass only) ----------------
#if defined(__AMDGCN__) && __has_builtin(__builtin_amdgcn_global_load_async_to_lds_b128)
#define PG_ASYNC 1
typedef __attribute__((address_space(1))) v4i* pg_gptr;
typedef __attribute__((address_space(3))) v4i* pg_lptr;
#else
#define PG_ASYNC 0
#if defined(__AMDGCN__)
#warning "PG: __builtin_amdgcn_global_load_async_to_lds_b128 NOT available; using sync LDS staging"
#endif
#endif

#if defined(__AMDGCN__) && __has_builtin(__builtin_amdgcn_wmma_f32_16x16x4_f32)
#define PG_WMMA 1
#else
#define PG_WMMA 0
#if defined(__AMDGCN__)
#warning "PG: __builtin_amdgcn_wmma_f32_16x16x4_f32 NOT available; using LDS-atomic group sums"
#endif
#endif

__device__ __forceinline__ void pg_wait_async0() {
#if defined(__AMDGCN__)
#if __has_builtin(__builtin_amdgcn_s_wait_asynccnt)
  __builtin_amdgcn_s_wait_asynccnt(0);
#else
  asm volatile("s_wait_asynccnt 0" ::: "memory");
#endif
#endif
}

// =====================================================================
// Kernel 0: per-batch sparse-structure setup (8 blocks x 256 threads).
// Target column of copy position j: tgt = idx<V ? idx : V+j (tgt==0
// dropped: onehot column 0 is zeroed in the reference). Collapse
// colliding targets into groups; emit gid[j] (-1 invalid), ngroups, and
// a dense packed vocab table gtab[v] = (gid2+1)<<16 | (gid1+1).
// =====================================================================
__global__ void pg_setup(const int* __restrict__ idx1, const int* __restrict__ idx2,
                         int* __restrict__ gid1, int* __restrict__ gid2,
                         int* __restrict__ ngr, int* __restrict__ gtab) {
  const int b = blockIdx.x;
  const int j = threadIdx.x;
  __shared__ int stg[kJ];
  __shared__ int slead[kJ];

  int tgtR[2], gidR[2], leadR[2];

  for (int cp = 0; cp < 2; ++cp) {
    const int* idx = cp ? idx2 : idx1;
    const int iv   = idx[b * kJ + j];
    const int tgt  = (iv < kV) ? iv : (kV + j);
    const bool valid = (tgt != 0);
    stg[j] = valid ? tgt : -1;
    __syncthreads();

    int j0 = j;
    if (valid) { j0 = 0; while (j0 < j && stg[j0] != tgt) ++j0; }
    const bool lead = valid && (j0 == j);
    slead[j] = lead ? 1 : 0;
    __syncthreads();

    int g = -1;
    if (valid) { int c = 0; for (int k = 0; k < j0; ++k) c += slead[k]; g = c; }
    if (j == 0) { int ng = 0; for (int k = 0; k < kJ; ++k) ng += slead[k]; ngr[b * 2 + cp] = ng; }
    (cp ? gid2 : gid1)[b * kJ + j] = g;

    tgtR[cp] = tgt; gidR[cp] = g; leadR[cp] = lead ? 1 : 0;
    __syncthreads();  // stg/slead reused next iteration
  }

  for (int v = j; v < kV; v += kJ) gtab[b * kV + v] = 0;
  __syncthreads();
  if (leadR[0] && tgtR[0] < kV) atomicOr(&gtab[b * kV + tgtR[0]], (gidR[0] + 1));
  if (leadR[1] && tgtR[1] < kV) atomicOr(&gtab[b * kV + tgtR[1]], (gidR[1] + 1) << 16);
}

// =====================================================================
// Kernel 1: group sums GS[b][t][g] = sum_{j: gid[j]==g} cp[b][t][j]
// == cp_score[b] (256x256) x M[b] (256x256 one-hot columns) as a dense
// f32 WMMA GEMM. Register-blocked: 4 accumulator tiles per K sweep so
// each A fragment feeds 4 WMMAs immediately (no hoisting, no spills).
// Grid: 16 blocks (b,cp) x 256 threads (8 wave32).
// =====================================================================
__global__ void __launch_bounds__(256) pg_groupsum(
    const float* __restrict__ cp1, const float* __restrict__ cp2,
    const int* __restrict__ gid1, const int* __restrict__ gid2,
    float* __restrict__ GS1, float* __restrict__ GS2) {
  const int b  = blockIdx.x >> 1;
  const int cp = blockIdx.x & 1;
  const float* A   = (cp ? cp2 : cp1) + (size_t)b * kT * kJ;   // [t][j]
  const int*   gid = (cp ? gid2 : gid1) + b * kJ;
  float*       GS  = (cp ? GS2 : GS1) + (size_t)b * kT * kJ;   // [t][g]

  __shared__ int sgid[kJ];
  sgid[threadIdx.x] = gid[threadIdx.x];
  __syncthreads();

#if PG_WMMA
  const int wave = threadIdx.x >> 5;
  const int lane = threadIdx.x & 31;
  const int lhi  = lane >> 4;   // 0/1: K half
  const int llo  = lane & 15;   // row / col within tile

  for (int tt = wave * 2; tt < wave * 2 + 2; ++tt) {   // 8 waves x 2 = 16 t-tiles
    const int tbase = tt * 16;
    const float* arow = A + (size_t)(tbase + llo) * kJ;  // this lane's A row
    for (int gc = 0; gc < 4; ++gc) {                   // 4 chunks of 4 g-tiles
      const int c0 = gc * 64 + llo;                    // lane's column in sub-tile 0
      v8f acc0 = {}, acc1 = {}, acc2 = {}, acc3 = {};
#pragma unroll 1
      for (int k0 = 0; k0 < kJ; k0 += 4) {
        const int ka = k0 + (lhi << 1);
        // A fragment (16x4): lane row = tbase+llo, K pair (ka, ka+1)
        v2f a; a.x = arow[ka]; a.y = arow[ka + 1];
        // B fragments: one-hot columns generated from group ids
        const int s0 = sgid[ka], s1 = sgid[ka + 1];
        v2f b0, b1, b2, b3;
        b0.x = (s0 == c0)      ? 1.0f : 0.0f;  b0.y = (s1 == c0)      ? 1.0f : 0.0f;
        b1.x = (s0 == c0 + 16) ? 1.0f : 0.0f;  b1.y = (s1 == c0 + 16) ? 1.0f : 0.0f;
        b2.x = (s0 == c0 + 32) ? 1.0f : 0.0f;  b2.y = (s1 == c0 + 32) ? 1.0f : 0.0f;
        b3.x = (s0 == c0 + 48) ? 1.0f : 0.0f;  b3.y = (s1 == c0 + 48) ? 1.0f : 0.0f;
        acc0 = __builtin_amdgcn_wmma_f32_16x16x4_f32(false, a, false, b0, (short)0, acc0, false, false);
        acc1 = __builtin_amdgcn_wmma_f32_16x16x4_f32(false, a, false, b1, (short)0, acc1, false, false);
        acc2 = __builtin_amdgcn_wmma_f32_16x16x4_f32(false, a, false, b2, (short)0, acc2, false, false);
        acc3 = __builtin_amdgcn_wmma_f32_16x16x4_f32(false, a, false, b3, (short)0, acc3, false, false);
      }
#pragma unroll
      for (int i = 0; i < 8; ++i) {   // D layout: VGPR i -> M = i + 8*lhi, N = lane col
        const size_t r = (size_t)(tbase + i + (lhi << 3)) * kJ;
        GS[r + c0]      = acc0[i];
        GS[r + c0 + 16] = acc1[i];
        GS[r + c0 + 32] = acc2[i];
        GS[r + c0 + 48] = acc3[i];
      }
    }
  }
#else
  __shared__ float sgs[kJ];
  for (int t = 0; t < kT; ++t) {
    sgs[threadIdx.x] = 0.0f;
    __syncthreads();
    const int g = sgid[threadIdx.x];
    if (g >= 0) atomicAdd(&sgs[g], A[(size_t)t * kJ + threadIdx.x]);
    __syncthreads();
    GS[(size_t)t * kJ + threadIdx.x] = sgs[threadIdx.x];
    __syncthreads();
  }
#endif
}

// =====================================================================
// Kernel 2: main streaming kernel. One block per (b,t) row.
// Async-stage gen row (64KB) into LDS, compute Z (caching exp(gen) back
// into the LDS row), then emit outputs.
// =====================================================================
__global__ void __launch_bounds__(256) pg_main(
    const float* __restrict__ gen, const float* __restrict__ cp1, const float* __restrict__ cp2,
    const int* __restrict__ idx1, const int* __restrict__ idx2,
    const int* __restrict__ ngr, const int* __restrict__ gtab,
    const float* __restrict__ GS1, const float* __restrict__ GS2,
    float* __restrict__ out) {
  const int row = blockIdx.x;   // b*kT + t
  const int b   = row >> 8;
  const int tid = threadIdx.x;

  __shared__ __align__(16) float gen_lds[kV];   // 64000 B (LDS is 320KB/WG on CDNA5)
  __shared__ float eg1[kJ], eg2[kJ], accv[kNOOV];
  __shared__ float red[8];
  __shared__ float s_logZ;

  // ---- 1) stream gen row into LDS (async DMA path on CDNA5) ----
  const float4* gsrc = (const float4*)(gen + (size_t)row * kV);
  for (int f = tid; f < kV / 4; f += 256) {
#if PG_ASYNC
    __builtin_amdgcn_global_load_async_to_lds_b128(
        (pg_gptr)(gsrc + f),
        (pg_lptr)(gen_lds + 4 * f),
        0, 0);
#else
    ((float4*)gen_lds)[f] = gsrc[f];
#endif
  }

  // ---- 2) overlap: copy-score side of Z from precomputed group sums ----
  const float mycp1 = cp1[(size_t)row * kJ + tid];
  const float mycp2 = cp2[(size_t)row * kJ + tid];
  const int ng1 = ngr[b * 2 + 0];
  const int ng2 = ngr[b * 2 + 1];
  float e1 = 0.0f, e2 = 0.0f;
  if (tid < ng1) e1 = __expf(GS1[(size_t)row * kJ + tid]);
  if (tid < ng2) e2 = __expf(GS2[(size_t)row * kJ + tid]);
  eg1[tid] = e1;
  eg2[tid] = e2;
  float part = e1 + e2;   // hit columns; un-hit columns contribute exp(0)=1 (added below)

  pg_wait_async0();
  __syncthreads();

  // ---- 3) Z_gen over the LDS-resident row; cache exp(gen) back in LDS ----
  float4* glv = (float4*)gen_lds;
  for (int f = tid; f < kV / 4; f += 256) {
    const float4 g = glv[f];
    float4 e;
    e.x = __expf(g.x); e.y = __expf(g.y); e.z = __expf(g.z); e.w = __expf(g.w);
    part += (e.x + e.y) + (e.z + e.w);
    glv[f] = e;   // same thread re-reads these in the emission pass
  }
  for (int o = 16; o > 0; o >>= 1) part += __shfl_down(part, o, 32);
  const int lane = tid & 31, wv = tid >> 5;
  if (lane == 0) red[wv] = part;
  __syncthreads();
  if (tid == 0) {
    float z = 0.0f;
    for (int i = 0; i < 8; ++i) z += red[i];
    z += (float)(2 * kVOOV - ng1 - ng2);   // un-hit copy columns: exp(0)=1 each
    s_logZ = __logf(z);
  }
  __syncthreads();
  const float logZ = s_logZ;

  // ---- 4) emit gen_final: log(e^gen + e^cps1 + e^cps2) - logZ ----
  float* orow = out + (size_t)row * kVOOV;
  const int4* gt4 = (const int4*)(gtab + b * kV);
  for (int f = tid; f < kV / 4; f += 256) {
    const float4 e = glv[f];   // exp(gen), cached by this thread in pass 3
    const int4 pk  = gt4[f];
    float4 o4;
    {
      float t1 = 1.0f, t2 = 1.0f; int h;
      h = pk.x & 0xFFFF;         if (h) t1 = eg1[h - 1];
      h = (pk.x >> 16) & 0xFFFF; if (h) t2 = eg2[h - 1];
      o4.x = __logf(e.x + t1 + t2) - logZ;
      t1 = 1.0f; t2 = 1.0f;
      h = pk.y & 0xFFFF;         if (h) t1 = eg1[h - 1];
      h = (pk.y >> 16) & 0xFFFF; if (h) t2 = eg2[h - 1];
      o4.y = __logf(e.y + t1 + t2) - logZ;
      t1 = 1.0f; t2 = 1.0f;
      h = pk.z & 0xFFFF;         if (h) t1 = eg1[h - 1];
      h = (pk.z >> 16) & 0xFFFF; if (h) t2 = eg2[h - 1];
      o4.z = __logf(e.z + t1 + t2) - logZ;
      t1 = 1.0f; t2 = 1.0f;
      h = pk.w & 0xFFFF;         if (h) t1 = eg1[h - 1];
      h = (pk.w >> 16) & 0xFFFF; if (h) t2 = eg2[h - 1];
      o4.w = __logf(e.w + t1 + t2) - logZ;
    }
    ((float4*)orow)[f] = o4;
  }

  // ---- 5) OOV buckets: scatter-LSE via LDS float atomics (ds_add_f32) ----
  accv[tid] = 0.0f;
  __syncthreads();
  const int i1 = idx1[b * kJ + tid];
  if (i1 >= kV) atomicAdd(&accv[i1 - kV], __expf(mycp1));
  const int i2 = idx2[b * kJ + tid];
  if (i2 >= kV) atomicAdd(&accv[i2 - kV], __expf(mycp2));
  __syncthreads();
  const float s = accv[tid];
  orow[kV + tid] = (s > 0.0f) ? (__logf(s) - logZ) : -1e20f;
}

// =====================================================================
extern "C" void kernel_launch(void* const* d_in, const int* in_sizes, int n_in,
                              void* d_out, int out_size, void* d_ws, size_t ws_size,
                              hipStream_t stream) {
  (void)in_sizes; (void)n_in; (void)out_size; (void)ws_size;
  const float* gen  = (const float*)d_in[0];
  const float* cp1  = (const float*)d_in[1];
  const float* cp2  = (const float*)d_in[2];
  const int*   idx1 = (const int*)d_in[5];
  const int*   idx2 = (const int*)d_in[6];

  // workspace layout (all 16B aligned): ~4.5 MB total
  char* ws   = (char*)d_ws;
  int*  gid1 = (int*)(ws + 0);                        // 8*256 ints   = 8192 B
  int*  gid2 = (int*)(ws + 8192);                     // 8192 B
  int*  ngr  = (int*)(ws + 16384);                    // 16 ints (pad 64 B)
  int*  gtab = (int*)(ws + 16448);                    // 8*16000 ints = 512000 B
  float* GS1 = (float*)(ws + 528448);                 // 8*256*256 f32 = 2 MB
  float* GS2 = (float*)(ws + 528448 + 2097152);       // 2 MB

  pg_setup<<<kB, kJ, 0, stream>>>(idx1, idx2, gid1, gid2, ngr, gtab);
  pg_groupsum<<<kB * 2, 256, 0, stream>>>(cp1, cp2, gid1, gid2, GS1, GS2);
  pg_main<<<kB * kT, 256, 0, stream>>>(gen, cp1, cp2, idx1, idx2, ngr, gtab,
                                       GS1, GS2, (float*)d_out);
}